// ZeroPoRF_10187662426217
// MI455X (gfx1250) — compile-verified
//
#include <hip/hip_runtime.h>
#include <hip/hip_bf16.h>

typedef float v2f __attribute__((ext_vector_type(2)));
typedef float v8f __attribute__((ext_vector_type(8)));

#define PLANE_W   512
#define PLANE_HW  (512 * 512)
#define NUM_CAMS  4096
#define N_RAYS    524288
#define HIDDEN    128
#define IN_FEATS  96
#define ALPHA_SC  1.0e-4f

#define CSTR 97    // codes LDS row stride (odd -> bank-conflict free)
#define HSTR 129   // hidden LDS row stride

__device__ __forceinline__ float silu_f(float v) {
    return v * (1.0f / (1.0f + __expf(-v)));
}
__device__ __forceinline__ int iclamp(int v, int lo, int hi) {
    return v < lo ? lo : (v > hi ? hi : v);
}

// One wave per 16-camera tile. Computes grid-sample features, the 3-layer MLP
// (layers 1-2 via V_WMMA_F32_16X16X4_F32), Rodrigues exp, and writes the
// per-camera 4x4 pose matrix to the pose table.
__global__ __launch_bounds__(32) void zeroporf_cam_pose_kernel(
    const float* __restrict__ t,
    const float* __restrict__ plane_xy,
    const float* __restrict__ plane_xz,
    const float* __restrict__ plane_yz,
    const float* __restrict__ w1, const float* __restrict__ b1,
    const float* __restrict__ w2, const float* __restrict__ b2,
    const float* __restrict__ w3, const float* __restrict__ b3,
    float* __restrict__ pose)
{
    __shared__ float s_codes[16 * CSTR];  // 16 cams x 96 features
    __shared__ float s_h1[16 * HSTR];     // hidden layer 1
    __shared__ float s_h2[16 * HSTR];     // hidden layer 2
    __shared__ int   s_off[3 * 16 * 4];   // bilinear texel offsets
    __shared__ float s_wt[3 * 16 * 4];    // bilinear weights
    __shared__ float s_t[16 * 3];         // camera translations
    __shared__ float s_rot[16 * 3];       // axis-angle outputs

    const int lane = threadIdx.x;   // 0..31
    const int m    = lane & 15;     // camera row within tile / N within tile
    const int hi   = lane >> 4;     // half-wave selector
    const int cam0 = blockIdx.x * 16;

    // ---- per-camera bilinear setup (lanes 0..15) -------------------------
    if (lane < 16) {
        const int cam = cam0 + m;
        const float tx = t[cam * 3 + 0];
        const float ty = t[cam * 3 + 1];
        const float tz = t[cam * 3 + 2];
        s_t[m * 3 + 0] = tx; s_t[m * 3 + 1] = ty; s_t[m * 3 + 2] = tz;
        // AXES = ((0,1),(0,2),(1,2)); coords[:,0]->W(x), coords[:,1]->H(y)
        const float cxs[3] = {tx, tx, ty};
        const float cys[3] = {ty, tz, tz};
        for (int p = 0; p < 3; ++p) {
            const float x  = (cxs[p] + 1.0f) * 0.5f * 511.0f;
            const float y  = (cys[p] + 1.0f) * 0.5f * 511.0f;
            const float xf = floorf(x), yf = floorf(y);
            const float wx = x - xf,   wy = y - yf;
            const int x0 = iclamp((int)xf,     0, 511);
            const int x1 = iclamp((int)xf + 1, 0, 511);
            const int y0 = iclamp((int)yf,     0, 511);
            const int y1 = iclamp((int)yf + 1, 0, 511);
            const int b = (p * 16 + m) * 4;
            s_off[b + 0] = y0 * PLANE_W + x0;
            s_off[b + 1] = y0 * PLANE_W + x1;
            s_off[b + 2] = y1 * PLANE_W + x0;
            s_off[b + 3] = y1 * PLANE_W + x1;
            s_wt[b + 0] = (1.f - wx) * (1.f - wy);
            s_wt[b + 1] = wx * (1.f - wy);
            s_wt[b + 2] = (1.f - wx) * wy;
            s_wt[b + 3] = wx * wy;
        }
    }
    __syncthreads();

    // ---- tri-plane gather: lane = channel (0..31) ------------------------
    const float* planes[3] = {plane_xy, plane_xz, plane_yz};
    for (int p = 0; p < 3; ++p) {
        const float* pl = planes[p] + (size_t)lane * PLANE_HW;
        for (int mm = 0; mm < 16; ++mm) {
            const int b = (p * 16 + mm) * 4;
            const float v = s_wt[b + 0] * pl[s_off[b + 0]]
                          + s_wt[b + 1] * pl[s_off[b + 1]]
                          + s_wt[b + 2] * pl[s_off[b + 2]]
                          + s_wt[b + 3] * pl[s_off[b + 3]];
            s_codes[mm * CSTR + p * 32 + lane] = v;
        }
    }
    __syncthreads();

    // ---- layer 1: (16x96) @ w1(96x128) + b1, SiLU ------------------------
    // A 16x4 f32 fragment: lanes 0-15 hold (K, K+1), lanes 16-31 hold (K+2, K+3)
    // for row M = lane&15. B 4x16 fragment mirrors the K striping.
    for (int nt = 0; nt < 8; ++nt) {
        const int n = nt * 16 + m;
        v8f acc = {};
        for (int k0 = 0; k0 < IN_FEATS; k0 += 4) {
            const int k = k0 + 2 * hi;
            v2f a, b;
            a.x = s_codes[m * CSTR + k];
            a.y = s_codes[m * CSTR + k + 1];
            b.x = w1[(size_t)k * HIDDEN + n];
            b.y = w1[(size_t)(k + 1) * HIDDEN + n];
            acc = __builtin_amdgcn_wmma_f32_16x16x4_f32(
                false, a, false, b, (short)0, acc, false, false);
        }
        const float bias = b1[n];
        for (int v = 0; v < 8; ++v) {
            const int row = v + 8 * hi;  // C layout: VGPR v -> M = v + 8*hi
            s_h1[row * HSTR + n] = silu_f(acc[v] + bias);
        }
    }
    __syncthreads();

    // ---- layer 2: (16x128) @ w2(128x128) + b2, SiLU ----------------------
    for (int nt = 0; nt < 8; ++nt) {
        const int n = nt * 16 + m;
        v8f acc = {};
        for (int k0 = 0; k0 < HIDDEN; k0 += 4) {
            const int k = k0 + 2 * hi;
            v2f a, b;
            a.x = s_h1[m * HSTR + k];
            a.y = s_h1[m * HSTR + k + 1];
            b.x = w2[(size_t)k * HIDDEN + n];
            b.y = w2[(size_t)(k + 1) * HIDDEN + n];
            acc = __builtin_amdgcn_wmma_f32_16x16x4_f32(
                false, a, false, b, (short)0, acc, false, false);
        }
        const float bias = b2[n];
        for (int v = 0; v < 8; ++v) {
            const int row = v + 8 * hi;
            s_h2[row * HSTR + n] = silu_f(acc[v] + bias);
        }
    }
    __syncthreads();

    // ---- layer 3: (16x128) @ w3(128x3) + b3, * ALPHA ---------------------
    for (int idx = lane; idx < 48; idx += 32) {
        const int mm = idx / 3;
        const int j  = idx - mm * 3;
        float sum = b3[j];
        #pragma unroll 8
        for (int k = 0; k < HIDDEN; ++k)
            sum += s_h2[mm * HSTR + k] * w3[(size_t)k * 3 + j];
        s_rot[idx] = sum * ALPHA_SC;
    }
    __syncthreads();

    // ---- Rodrigues + pose store (lanes 0..15) ----------------------------
    if (lane < 16) {
        const int cam = cam0 + m;
        const float r0 = s_rot[m * 3 + 0];
        const float r1 = s_rot[m * 3 + 1];
        const float r2 = s_rot[m * 3 + 2];
        const float dot = r0 * r0 + r1 * r1 + r2 * r2;
        const float th  = sqrtf(dot);
        const float Af  = sinf(th) / (th + 1e-10f);
        const float Bf  = (1.0f - cosf(th)) / (dot + 1e-10f);
        const float r[3] = {r0, r1, r2};
        const float K[3][3] = {{0.f, -r2, r1}, {r2, 0.f, -r0}, {-r1, r0, 0.f}};
        float* P = pose + (size_t)cam * 16;
        #pragma unroll
        for (int i = 0; i < 3; ++i) {
            #pragma unroll
            for (int j = 0; j < 3; ++j) {
                // R = I + Af*K + Bf*(K@K),  K@K = r r^T - dot * I
                float v = Af * K[i][j] + Bf * r[i] * r[j];
                if (i == j) v += 1.0f - Bf * dot;
                P[i * 4 + j] = v;
            }
            P[i * 4 + 3] = s_t[m * 3 + i];
        }
        P[12] = 0.0f; P[13] = 0.0f; P[14] = 0.0f; P[15] = 1.0f;
    }
}

// One thread per output element: out[ray] = pose[cam] @ init_c2w[cam].
// Fully coalesced 32 MB store; both 256 KB tables stay L2-resident.
__global__ __launch_bounds__(256) void zeroporf_compose_kernel(
    const int*   __restrict__ cam_id,
    const float* __restrict__ pose,
    const float* __restrict__ init_c2w,
    float*       __restrict__ out)
{
    const int idx = blockIdx.x * 256 + threadIdx.x;  // 0 .. N_RAYS*16-1
    const int ray = idx >> 4;
    const int e   = idx & 15;
    const int i   = e >> 2;
    const int j   = e & 3;
    const int cam = cam_id[ray];
    const float* A = pose     + (size_t)cam * 16 + i * 4;
    const float* B = init_c2w + (size_t)cam * 16 + j;
    out[idx] = A[0] * B[0] + A[1] * B[4] + A[2] * B[8] + A[3] * B[12];
}

extern "C" void kernel_launch(void* const* d_in, const int* in_sizes, int n_in,
                              void* d_out, int out_size, void* d_ws, size_t ws_size,
                              hipStream_t stream) {
    // setup_inputs order:
    // 0 cam_id(i32) 1 t 2 plane_xy 3 plane_xz 4 plane_yz
    // 5 w1 6 b1 7 w2 8 b2 9 w3 10 b3 11 init_c2w
    const int*   cam_id   = (const int*)  d_in[0];
    const float* t        = (const float*)d_in[1];
    const float* plane_xy = (const float*)d_in[2];
    const float* plane_xz = (const float*)d_in[3];
    const float* plane_yz = (const float*)d_in[4];
    const float* w1       = (const float*)d_in[5];
    const float* b1       = (const float*)d_in[6];
    const float* w2       = (const float*)d_in[7];
    const float* b2       = (const float*)d_in[8];
    const float* w3       = (const float*)d_in[9];
    const float* b3       = (const float*)d_in[10];
    const float* init_c2w = (const float*)d_in[11];
    float* out  = (float*)d_out;
    float* pose = (float*)d_ws;  // NUM_CAMS * 16 floats = 256 KB

    zeroporf_cam_pose_kernel<<<NUM_CAMS / 16, 32, 0, stream>>>(
        t, plane_xy, plane_xz, plane_yz, w1, b1, w2, b2, w3, b3, pose);

    zeroporf_compose_kernel<<<(N_RAYS * 16) / 256, 256, 0, stream>>>(
        cam_id, pose, init_c2w, out);
}